// MovingAverageLayer_6554120093996
// MI455X (gfx1250) — compile-verified
//
#include <hip/hip_runtime.h>
#include <hip/hip_bf16.h>

// MovingAverage: out[T,256] = concat(x, MA7, MA30, MA90) over rows of x[T,64].
// Bandwidth-bound (~320MB traffic, ~14us at 23.3TB/s). Strategy:
//  - one block per 256-row chunk, 96-row halo (max period 90) -> no global scan
//  - async global->LDS b128 loads (CDNA5 async path, ASYNCcnt)
//  - local segmented prefix-sum in LDS (16 segments x float4 feature groups)
//  - outputs via non-temporal b128 stores (write-once 256MB, keep L2 for input)

#define CHUNK   256
#define HALO    96
#define ROWS    (CHUNK + HALO)     // 352
#define NF      64
#define NSEG    16
#define SEGROWS (ROWS / NSEG)      // 22
#define NTHREADS 256

typedef float v4f __attribute__((ext_vector_type(4)));   // native clang vector: b128 + NT-store legal

__global__ __launch_bounds__(NTHREADS)
void ma_scan_kernel(const float* __restrict__ x, float* __restrict__ out) {
    // 352*64*4 = 90112B cumsum, 256*64*4 = 65536B raw copy, 16*64*4 = 4096B totals
    __shared__ float cs[ROWS * NF];
    __shared__ float xraw[CHUNK * NF];
    __shared__ float totals[NSEG * NF];

    const int t   = threadIdx.x;
    const int blk = blockIdx.x;
    const int chunkStart = blk * CHUNK;

    // ---------------- Phase 0: global -> LDS (async b128) ----------------
    const unsigned lds_cs = (unsigned)(size_t)&cs[0];   // low 32 bits = LDS byte offset
    if (blk == 0) {
        // zero the halo (acts as cumsum[-1]=0 so local cumsum == global cumsum)
        v4f z = 0.0f;
        for (int k = 0; k < (HALO * NF) / (4 * NTHREADS); ++k)      // 6
            *(v4f*)&cs[(k * NTHREADS + t) * 4] = z;
        const float* gsrc = x;                                       // row 0
        const unsigned lbase = lds_cs + HALO * NF * 4;
        for (int k = 0; k < (CHUNK * NF * 4) / (NTHREADS * 16); ++k) { // 16
            unsigned off   = (unsigned)(t * 16 + k * 4096);
            unsigned laddr = lbase + off;
            asm volatile("global_load_async_to_lds_b128 %0, %1, %2"
                         :: "v"(laddr), "v"(off), "s"(gsrc) : "memory");
        }
    } else {
        const float* gsrc = x + (size_t)(chunkStart - HALO) * NF;
        for (int k = 0; k < (ROWS * NF * 4) / (NTHREADS * 16); ++k) {  // 22
            unsigned off   = (unsigned)(t * 16 + k * 4096);
            unsigned laddr = lds_cs + off;
            asm volatile("global_load_async_to_lds_b128 %0, %1, %2"
                         :: "v"(laddr), "v"(off), "s"(gsrc) : "memory");
        }
    }
    asm volatile("s_wait_asynccnt 0" ::: "memory");
    __syncthreads();

    // ---------------- Phase 1: segmented prefix sum over rows ----------------
    const int fg  = t & (NSEG - 1);   // feature group: 4 consecutive features
    const int seg = t >> 4;           // row segment 0..15 (22 rows each)
    const int fb  = fg * 4;
    const int r0  = seg * SEGROWS;

    v4f run = 0.0f;
    #pragma unroll
    for (int j = 0; j < SEGROWS; ++j) {
        const int r = r0 + j;
        v4f v = *(v4f*)&cs[r * NF + fb];
        if (r >= HALO)                                  // keep exact x for passthrough
            *(v4f*)&xraw[(r - HALO) * NF + fb] = v;
        run += v;
        *(v4f*)&cs[r * NF + fb] = run;
    }
    *(v4f*)&totals[seg * NF + fb] = run;
    __syncthreads();

    v4f carry = 0.0f;
    for (int s = 0; s < seg; ++s)
        carry += *(v4f*)&totals[s * NF + fb];
    if (seg > 0) {
        #pragma unroll
        for (int j = 0; j < SEGROWS; ++j) {
            const int r = r0 + j;
            *(v4f*)&cs[r * NF + fb] += carry;
        }
    }
    __syncthreads();

    // ---------------- Phase 2: windowed averages + NT streaming stores ----------------
    const int rg = t >> 4;            // 16 row groups, rows stride 16
    #pragma unroll 4
    for (int k = 0; k < CHUNK / NSEG; ++k) {            // 16 rows/thread
        const int lr = rg + k * NSEG;                   // local chunk row
        const int li = lr + HALO;                       // index into cs
        const int gi = chunkStart + lr;                 // global row

        v4f c0  = *(v4f*)&cs[(li)      * NF + fb];
        v4f c7  = *(v4f*)&cs[(li - 7)  * NF + fb];
        v4f c30 = *(v4f*)&cs[(li - 30) * NF + fb];
        v4f c90 = *(v4f*)&cs[(li - 90) * NF + fb];
        v4f xv  = *(v4f*)&xraw[lr * NF + fb];

        v4f m7  = (c0 - c7)  * (1.0f / 7.0f);
        v4f m30 = (c0 - c30) * (1.0f / 30.0f);
        v4f m90 = (c0 - c90) * (1.0f / 90.0f);

        if (blk == 0) {  // expanding mean for global rows < p-1 (only first chunk)
            const float inv = 1.0f / (float)(gi + 1);
            if (gi < 6)  m7  = c0 * inv;
            if (gi < 29) m30 = c0 * inv;
            if (gi < 89) m90 = c0 * inv;
        }

        float* orow = out + (size_t)gi * (4 * NF);
        __builtin_nontemporal_store(xv,  (v4f*)(orow + fb));
        __builtin_nontemporal_store(m7,  (v4f*)(orow + NF + fb));
        __builtin_nontemporal_store(m30, (v4f*)(orow + 2 * NF + fb));
        __builtin_nontemporal_store(m90, (v4f*)(orow + 3 * NF + fb));
    }
}

extern "C" void kernel_launch(void* const* d_in, const int* in_sizes, int n_in,
                              void* d_out, int out_size, void* d_ws, size_t ws_size,
                              hipStream_t stream) {
    (void)n_in; (void)d_ws; (void)ws_size; (void)out_size;
    const float* x = (const float*)d_in[0];
    float* out = (float*)d_out;
    const int T = in_sizes[0] / NF;       // 262144
    const int nblocks = T / CHUNK;        // 1024 (T divides evenly)
    ma_scan_kernel<<<dim3(nblocks), dim3(NTHREADS), 0, stream>>>(x, out);
}